// pyg_GCNLayer_2010044694725
// MI455X (gfx1250) — compile-verified
//
#include <hip/hip_runtime.h>

// CDNA5 / gfx1250 GCN layer:
//   agg = scatter_add(feats[src] -> dst)                       (atomic f32, L2-resident)
//   out = relu(agg@Wrel^T + brel + feats@Wroot^T) + relu(feats@Wres^T + bres)
//   y   = batchnorm(out) * gamma + beta
// GEMMs use v_wmma_f32_16x16x4_f32 (fp32 matrix pipe, exact precision).

typedef __attribute__((ext_vector_type(2))) float v2f;
typedef __attribute__((ext_vector_type(8))) float v8f;

#define DIN   128
#define DOUT  128
#define BN_EPS 1e-5f

// ---------------------------------------------------------------- zero ws ---
__global__ void k_zero(float* __restrict__ p, int n) {
    int i = blockIdx.x * blockDim.x + threadIdx.x;
    if (i < n) p[i] = 0.0f;
}

// ------------------------------------------------------------- scatter-add --
// One wave32 per edge; lane l handles 4 consecutive floats (float4 gather,
// 4 hardware global_atomic_add_f32 scatters). 8 edges per 256-thread block.
__global__ __launch_bounds__(256) void k_scatter(const float* __restrict__ feats,
                                                 const long long* __restrict__ ei,
                                                 float* __restrict__ agg, int E) {
    int lane = threadIdx.x & 31;
    int e = blockIdx.x * (blockDim.x >> 5) + (threadIdx.x >> 5);
    if (e >= E) return;
    long long s = ei[e];                 // src row
    long long d = ei[(long long)E + e];  // dst row
    float4 v = ((const float4*)(feats + (size_t)s * DIN))[lane];
    float* dst = agg + (size_t)d * DIN + lane * 4;
    unsafeAtomicAdd(dst + 0, v.x);
    unsafeAtomicAdd(dst + 1, v.y);
    unsafeAtomicAdd(dst + 2, v.z);
    unsafeAtomicAdd(dst + 3, v.w);
}

// ----------------------------------------------- fused triple-GEMM + epilogue
// Block = 256 threads = 8 waves. Block owns 16 node rows; wave w owns output
// columns [16w, 16w+16). K=128 swept in steps of 4 -> 32 wmma per matrix,
// 96 wmma per wave. A tiles staged in LDS (shared by all 8 waves).
__global__ __launch_bounds__(256) void k_gemm_fused(
    const float* __restrict__ agg,  const float* __restrict__ feats,
    const float* __restrict__ Wrel, const float* __restrict__ brel,
    const float* __restrict__ Wroot,
    const float* __restrict__ Wres, const float* __restrict__ bres,
    float* __restrict__ newf,       // staging (d_out)
    float* __restrict__ colsum, float* __restrict__ colsumsq) {

    __shared__ float lds_a[16 * DIN];   // agg tile   (8 KB)
    __shared__ float lds_f[16 * DIN];   // feats tile (8 KB)

    const int row0 = blockIdx.x * 16;

    // Stage both 16x128 A tiles with float4 loads (2 per thread each).
    const float4* ga = (const float4*)(agg   + (size_t)row0 * DIN);
    const float4* gf = (const float4*)(feats + (size_t)row0 * DIN);
    for (int i = threadIdx.x; i < 16 * DIN / 4; i += 256) {
        ((float4*)lds_a)[i] = ga[i];
        ((float4*)lds_f)[i] = gf[i];
    }
    __syncthreads();

    const int lane = threadIdx.x & 31;
    const int wave = threadIdx.x >> 5;
    const int n0   = wave * 16;
    const int m    = lane & 15;          // A row / B,C column within tile
    const int kh   = (lane >> 4) * 2;    // K sub-offset: 0 (lanes 0-15), 2 (16-31)
    const int ncol = n0 + m;             // this lane's global output column

    const float* wrel_row  = Wrel  + (size_t)ncol * DIN;  // B[k][n] = W[n][k]
    const float* wroot_row = Wroot + (size_t)ncol * DIN;
    const float* wres_row  = Wres  + (size_t)ncol * DIN;

    v8f acc_rel = {}, acc_root = {}, acc_res = {};

    #pragma unroll 8
    for (int k0 = 0; k0 < DIN; k0 += 4) {
        const int ka = k0 + kh;
        v2f a_agg, a_ft, b_rel, b_root, b_res;
        a_agg.x = lds_a[m * DIN + ka];   a_agg.y = lds_a[m * DIN + ka + 1];
        a_ft.x  = lds_f[m * DIN + ka];   a_ft.y  = lds_f[m * DIN + ka + 1];
        b_rel.x  = wrel_row[ka];         b_rel.y  = wrel_row[ka + 1];
        b_root.x = wroot_row[ka];        b_root.y = wroot_row[ka + 1];
        b_res.x  = wres_row[ka];         b_res.y  = wres_row[ka + 1];

        acc_rel  = __builtin_amdgcn_wmma_f32_16x16x4_f32(
                       false, a_agg, false, b_rel,  (short)0, acc_rel,  false, false);
        acc_root = __builtin_amdgcn_wmma_f32_16x16x4_f32(
                       false, a_ft,  false, b_root, (short)0, acc_root, false, false);
        acc_res  = __builtin_amdgcn_wmma_f32_16x16x4_f32(
                       false, a_ft,  false, b_res,  (short)0, acc_res,  false, false);
    }

    // Epilogue. C/D layout: VGPR r -> row (r + 8*(lane>>4)), lane&15 -> column.
    const float bc_rel = brel[ncol];
    const float bc_res = bres[ncol];
    const int   half   = lane >> 4;
    float ls = 0.0f, lsq = 0.0f;
    #pragma unroll
    for (int r = 0; r < 8; ++r) {
        const int row = row0 + half * 8 + r;
        float v = fmaxf(acc_rel[r] + bc_rel + acc_root[r], 0.0f)
                + fmaxf(acc_res[r] + bc_res,               0.0f);
        newf[(size_t)row * DOUT + ncol] = v;
        ls  += v;
        lsq += v * v;
    }
    // All 8 values of a lane share one column -> 2 atomics per lane.
    unsafeAtomicAdd(&colsum[ncol],   ls);
    unsafeAtomicAdd(&colsumsq[ncol], lsq);
}

// ----------------------------------------------------------- batchnorm pass -
__global__ void k_bn(float* __restrict__ out,
                     const float* __restrict__ colsum,
                     const float* __restrict__ colsumsq,
                     const float* __restrict__ gamma,
                     const float* __restrict__ beta,
                     int total, float invN) {
    int i = blockIdx.x * blockDim.x + threadIdx.x;
    if (i >= total) return;
    int j = i & (DOUT - 1);
    float mean = colsum[j] * invN;
    float var  = colsumsq[j] * invN - mean * mean;
    float x = out[i];
    out[i] = (x - mean) * rsqrtf(var + BN_EPS) * gamma[j] + beta[j];
}

// ------------------------------------------------------------------ launch --
extern "C" void kernel_launch(void* const* d_in, const int* in_sizes, int n_in,
                              void* d_out, int out_size, void* d_ws, size_t ws_size,
                              hipStream_t stream) {
    const float*     feats = (const float*)d_in[0];
    const long long* ei    = (const long long*)d_in[1];   // int64 edge_index [2,E]
    const float*     Wrel  = (const float*)d_in[2];
    const float*     brel  = (const float*)d_in[3];
    const float*     Wroot = (const float*)d_in[4];
    const float*     Wres  = (const float*)d_in[5];
    const float*     bres  = (const float*)d_in[6];
    const float*     gamma = (const float*)d_in[7];
    const float*     beta  = (const float*)d_in[8];
    float*           out   = (float*)d_out;

    const int N = in_sizes[0] / DIN;   // 100000 (multiple of 16)
    const int E = in_sizes[1] / 2;     // 1600000

    // Workspace: agg[N*DIN] | colsum[128] | colsumsq[128]
    float* agg      = (float*)d_ws;
    float* colsum   = agg + (size_t)N * DIN;
    float* colsumsq = colsum + DOUT;

    const int zcount = N * DIN + 2 * DOUT;
    k_zero<<<(zcount + 255) / 256, 256, 0, stream>>>(agg, zcount);

    k_scatter<<<(E + 7) / 8, 256, 0, stream>>>(feats, ei, agg, E);

    k_gemm_fused<<<(N + 15) / 16, 256, 0, stream>>>(
        agg, feats, Wrel, brel, Wroot, Wres, bres, out, colsum, colsumsq);

    const int total = N * DOUT;
    k_bn<<<(total + 255) / 256, 256, 0, stream>>>(
        out, colsum, colsumsq, gamma, beta, total, 1.0f / (float)N);
}